// MAE_Loss_27650999452083
// MI455X (gfx1250) — compile-verified
//
#include <hip/hip_runtime.h>
#include <math.h>

typedef __attribute__((ext_vector_type(2))) float v2f;
typedef __attribute__((ext_vector_type(8))) float v8f;

#define BATCH   4
#define NPTS    8192
#define CHUNK   256          // reference points staged in LDS per block iteration
#define WAVES   8            // waves per block (wave32)
#define TPB     (WAVES * 32) // 256 threads

// One wave owns a 16-row query tile. Distance bilinear form folded into the
// K=4 WMMA:  A row = (qx,qy,qz,q2),  B col = (-2rx,-2ry,-2rz,1)
//   => D = q2 - 2*q.r ;  d = D + r2  (pk_add epilogue), min/argmin; clamp at end.
__global__ __launch_bounds__(TPB) void dcd_nn_kernel(const float* __restrict__ q,
                                                     const float* __restrict__ r,
                                                     float* __restrict__ dmin,
                                                     int* __restrict__ idxmin)
{
    // B fragments pre-split per wave half:
    //   lanes 0-15 : VGPR0 = K0 (-2x), VGPR1 = K2 (-2z)   -> bLo
    //   lanes 16-31: VGPR0 = K1 (-2y), VGPR1 = K3 ( 1 )   -> bHi
    __shared__ v2f   bLo[CHUNK];
    __shared__ v2f   bHi[CHUNK];
    __shared__ float r2s[CHUNK];

    const int  tid  = threadIdx.x;
    const int  wave = tid >> 5;
    const int  lane = tid & 31;
    const int  col  = lane & 15;
    const bool hi   = lane >= 16;

    const int batch   = blockIdx.y;
    const int rowTile = blockIdx.x * WAVES + wave;     // tile of 16 query rows
    const int qrow    = rowTile * 16 + col;            // this lane's query row

    const float* qb = q + (size_t)(batch * NPTS) * 3;
    const float* rb = r + (size_t)(batch * NPTS) * 3;

    const float qx = qb[qrow * 3 + 0];
    const float qy = qb[qrow * 3 + 1];
    const float qz = qb[qrow * 3 + 2];
    const float q2 = qx * qx + qy * qy + qz * qz;

    // A fragment (16x4): lanes 0-15: {K0=qx, K1=qy}; lanes 16-31: {K2=qz, K3=q2}
    v2f afrag;
    afrag.x = hi ? qz : qx;
    afrag.y = hi ? q2 : qy;

    const v2f* __restrict__ bsel = hi ? bHi : bLo;   // loop-invariant select

    float bestD[8];
    int   bestI[8];
#pragma unroll
    for (int p = 0; p < 8; ++p) { bestD[p] = 3.4e38f; bestI[p] = 0; }

    for (int base = 0; base < NPTS; base += CHUNK) {
        __syncthreads();
        {   // stage CHUNK reference points (CHUNK == TPB: one point per thread)
            const int   j = tid;
            const float x = rb[(base + j) * 3 + 0];
            const float y = rb[(base + j) * 3 + 1];
            const float z = rb[(base + j) * 3 + 2];
            v2f lo;  lo.x  = -2.0f * x; lo.y  = -2.0f * z;
            v2f hi2; hi2.x = -2.0f * y; hi2.y = 1.0f;
            bLo[j] = lo;
            bHi[j] = hi2;
            r2s[j] = x * x + y * y + z * z;
        }
        __syncthreads();

#pragma unroll 4
        for (int t = 0; t < CHUNK / 16; ++t) {
            const int   j     = t * 16 + col;
            const v2f   bfrag = bsel[j];     // ds_load_b64 (pair-fused by compiler)
            const float r2l   = r2s[j];      // ds_load_b32

            v8f acc = {};
            // (neg_a, A, neg_b, B, c_mod, C, reuse_a, reuse_b)
            acc = __builtin_amdgcn_wmma_f32_16x16x4_f32(
                false, afrag, false, bfrag, (short)0, acc, false, false);

            // Phase 1: distances (contiguous regs + broadcast -> v_pk_add_f32 x4)
            float dv[8];
#pragma unroll
            for (int p = 0; p < 8; ++p) dv[p] = acc[p] + r2l;

            // Phase 2: min/argmin selects (clamp deferred to the end)
            const int jg = base + j; // local index within batch
#pragma unroll
            for (int p = 0; p < 8; ++p) {
                const bool take = dv[p] < bestD[p];
                bestD[p] = take ? dv[p] : bestD[p];
                bestI[p] = take ? jg    : bestI[p];
            }
        }
    }

    // Min/argmin across the 16 columns held by each half of the wave.
#pragma unroll
    for (int m = 1; m <= 8; m <<= 1) {
#pragma unroll
        for (int p = 0; p < 8; ++p) {
            const float od = __shfl_xor(bestD[p], m, 32);
            const int   oi = __shfl_xor(bestI[p], m, 32);
            const bool take = (od < bestD[p]) || (od == bestD[p] && oi < bestI[p]);
            bestD[p] = take ? od : bestD[p];
            bestI[p] = take ? oi : bestI[p];
        }
    }

    if (col == 0) {
        const int rbase = batch * NPTS + rowTile * 16 + (hi ? 8 : 0);
#pragma unroll
        for (int p = 0; p < 8; ++p) {
            dmin[rbase + p]   = fmaxf(bestD[p], 0.0f);  // clamp once here
            idxmin[rbase + p] = bestI[p];
        }
    }
}

// Occupancy counts: cnt[b, j] = #queries whose NN is j (deterministic int atomics).
__global__ __launch_bounds__(256) void dcd_count_kernel(const int* __restrict__ idx,
                                                        unsigned* __restrict__ cnt)
{
    const int g = blockIdx.x * blockDim.x + threadIdx.x;
    if (g < BATCH * NPTS) {
        const int b = g / NPTS;
        atomicAdd(&cnt[b * NPTS + idx[g]], 1u);
    }
}

// Single-block deterministic reduction:
// loss = 0.5 * mean(1 - exp(-d1)/w1) + 0.5 * mean(1 - exp(-d2)/w2)
__global__ __launch_bounds__(512) void dcd_final_kernel(const float* __restrict__ d1,
                                                        const int* __restrict__ i1,
                                                        const unsigned* __restrict__ cntT,
                                                        const float* __restrict__ d2,
                                                        const int* __restrict__ i2,
                                                        const unsigned* __restrict__ cntP,
                                                        float* __restrict__ out)
{
    __shared__ float red[512];
    float s = 0.0f;
    for (int g = threadIdx.x; g < BATCH * NPTS; g += 512) {
        const int b = g / NPTS;
        const float w1 = (float)cntT[b * NPTS + i1[g]];
        s += 1.0f - __expf(-d1[g]) / w1;
        const float w2 = (float)cntP[b * NPTS + i2[g]];
        s += 1.0f - __expf(-d2[g]) / w2;
    }
    red[threadIdx.x] = s;
    __syncthreads();
    for (int off = 256; off > 0; off >>= 1) {
        if (threadIdx.x < off) red[threadIdx.x] += red[threadIdx.x + off];
        __syncthreads();
    }
    if (threadIdx.x == 0) out[0] = 0.5f * red[0] / (float)(BATCH * NPTS);
}

extern "C" void kernel_launch(void* const* d_in, const int* in_sizes, int n_in,
                              void* d_out, int out_size, void* d_ws, size_t ws_size,
                              hipStream_t stream)
{
    (void)in_sizes; (void)n_in; (void)out_size; (void)ws_size;

    const float* pred = (const float*)d_in[0];
    const float* targ = (const float*)d_in[1];
    float* out = (float*)d_out;

    const int total = BATCH * NPTS;

    // Workspace layout (768 KB total)
    float*    d1   = (float*)d_ws;
    float*    d2   = d1 + total;
    int*      i1   = (int*)(d2 + total);
    int*      i2   = i1 + total;
    unsigned* cntT = (unsigned*)(i2 + total);
    unsigned* cntP = cntT + total;

    hipMemsetAsync(cntT, 0, 2 * (size_t)total * sizeof(unsigned), stream);

    dim3 grid(NPTS / 16 / WAVES, BATCH); // (64, 4)
    dcd_nn_kernel<<<grid, TPB, 0, stream>>>(pred, targ, d1, i1); // pred -> target
    dcd_nn_kernel<<<grid, TPB, 0, stream>>>(targ, pred, d2, i2); // target -> pred

    dcd_count_kernel<<<(total + 255) / 256, 256, 0, stream>>>(i1, cntT);
    dcd_count_kernel<<<(total + 255) / 256, 256, 0, stream>>>(i2, cntP);

    dcd_final_kernel<<<1, 512, 0, stream>>>(d1, i1, cntT, d2, i2, cntP, out);
}